// ContrastiveLoss_20512763806331
// MI455X (gfx1250) — compile-verified
//
#include <hip/hip_runtime.h>
#include <math.h>

#define N_DATA   500000
#define FEAT_DIM 128
#define BATCH    512
#define N_NEG    4096
#define INV_T    (1.0f / 0.07f)
#define EPS_N    1e-12f

typedef __attribute__((ext_vector_type(2))) float v2f;
typedef __attribute__((ext_vector_type(8))) float v8f;

// ---------------------------------------------------------------------------
// Kernel A: L2-normalize student & teacher rows, positive logits.
// 512 blocks x 128 threads (one thread per feature).
// ---------------------------------------------------------------------------
__global__ void normalize_kernel(const float* __restrict__ sf,
                                 const float* __restrict__ tf,
                                 float* __restrict__ s_n,
                                 float* __restrict__ t_n,
                                 float* __restrict__ pos) {
    const int b = blockIdx.x;
    const int t = threadIdx.x;
    __shared__ float r1[FEAT_DIM];
    __shared__ float r2[FEAT_DIM];

    float sv = sf[b * FEAT_DIM + t];
    float tv = tf[b * FEAT_DIM + t];

    r1[t] = sv * sv;
    r2[t] = tv * tv;
    __syncthreads();
    for (int o = FEAT_DIM / 2; o > 0; o >>= 1) {
        if (t < o) { r1[t] += r1[t + o]; r2[t] += r2[t + o]; }
        __syncthreads();
    }
    const float ns = sqrtf(r1[0]);
    const float nt = sqrtf(r2[0]);
    __syncthreads();

    const float sn = sv / fmaxf(ns, EPS_N);
    const float tn = tv / fmaxf(nt, EPS_N);
    s_n[b * FEAT_DIM + t] = sn;
    t_n[b * FEAT_DIM + t] = tn;

    r1[t] = sn * tn;
    __syncthreads();
    for (int o = FEAT_DIM / 2; o > 0; o >>= 1) {
        if (t < o) r1[t] += r1[t + o];
        __syncthreads();
    }
    if (t == 0) pos[b] = r1[0] * INV_T;
}

// ---------------------------------------------------------------------------
// Kernel B: gathered negative logits via V_WMMA_F32_16X16X4_F32.
// One block per batch row b, 8 waves; each wave owns 16-negative tiles.
// A operand = 16 gathered bank rows, loaded per-lane as b64 directly in the
// native 16x4 f32 A layout (lane m -> row m, lane halves -> K pairs).
// B operand = s[b] K-slice broadcast to all 16 columns from LDS.
// Accumulate K=128 in 32 WMMA steps, f32 end-to-end.
// ---------------------------------------------------------------------------
__global__ void __launch_bounds__(256)
neg_logits_kernel(const float* __restrict__ bank,
                  const int* __restrict__ neg_idx,
                  const float* __restrict__ s_n,
                  float* __restrict__ neg_logits) {
    __shared__ float s_row[FEAT_DIM];
    const int b = blockIdx.x;

    for (int i = threadIdx.x; i < FEAT_DIM; i += blockDim.x)
        s_row[i] = s_n[b * FEAT_DIM + i];
    __syncthreads();

    const int lane   = threadIdx.x & 31;
    const int wave   = threadIdx.x >> 5;
    const int nwaves = blockDim.x >> 5;
    const int m = lane & 15;   // matrix row within the 16-neg tile
    const int h = lane >> 4;   // A: 0 -> K {d,d+1}, 1 -> K {d+2,d+3}; B: 0 -> K {d,d+2}, 1 -> K {d+1,d+3}

    const int ntiles = N_NEG / 16;
    for (int tile = wave; tile < ntiles; tile += nwaves) {
        const int k0 = tile * 16;
        const int idx = neg_idx[b * N_NEG + k0 + m];
        const float* row = bank + (long long)idx * FEAT_DIM;

        v8f c = {};
#pragma unroll
        for (int kk = 0; kk < FEAT_DIM / 4; ++kk) {
            const int d0 = kk * 4;
            // A: global_load_b64 lands exactly in 16x4 f32 A-operand layout
            v2f a = *(const v2f*)(row + d0 + 2 * h);
            // B: s[b] contraction slice, replicated across the 16 columns
            v2f bb;
            bb[0] = s_row[d0 + h];
            bb[1] = s_row[d0 + h + 2];
            c = __builtin_amdgcn_wmma_f32_16x16x4_f32(
                    /*neg_a=*/false, a, /*neg_b=*/false, bb,
                    /*c_mod=*/(short)0, c, /*reuse_a=*/false, /*reuse_b=*/false);
        }

        // D column 0: accumulator VGPR g holds y[g] in lane 0, y[8+g] in lane 16
        float* outp = neg_logits + (long long)b * N_NEG + k0;
#pragma unroll
        for (int g = 0; g < 8; ++g) {
            const float v = c[g] * INV_T;
            if (lane == 0)  outp[g]     = v;
            if (lane == 16) outp[8 + g] = v;
        }
    }
}

// ---------------------------------------------------------------------------
// Kernel C1: per-row logsumexp([pos, neg...]) - pos.  512 blocks x 256 thr.
// ---------------------------------------------------------------------------
__global__ void lse_kernel(const float* __restrict__ neg_logits,
                           const float* __restrict__ pos,
                           float* __restrict__ row_out) {
    const int b = blockIdx.x;
    const int t = threadIdx.x;
    const float* row = neg_logits + (long long)b * N_NEG;
    const float p = pos[b];
    __shared__ float red[256];

    float mx = (t == 0) ? p : -INFINITY;
    for (int i = t; i < N_NEG; i += 256) mx = fmaxf(mx, row[i]);
    red[t] = mx;
    __syncthreads();
    for (int o = 128; o > 0; o >>= 1) {
        if (t < o) red[t] = fmaxf(red[t], red[t + o]);
        __syncthreads();
    }
    const float M = red[0];
    __syncthreads();

    float s = (t == 0) ? __expf(p - M) : 0.0f;
    for (int i = t; i < N_NEG; i += 256) s += __expf(row[i] - M);
    red[t] = s;
    __syncthreads();
    for (int o = 128; o > 0; o >>= 1) {
        if (t < o) red[t] += red[t + o];
        __syncthreads();
    }
    if (t == 0) row_out[b] = M + __logf(red[0]) - p;
}

// ---------------------------------------------------------------------------
// Kernel C2: mean over 512 rows -> d_out[0].  1 block x 512 threads.
// ---------------------------------------------------------------------------
__global__ void mean_kernel(const float* __restrict__ row_lse,
                            float* __restrict__ out) {
    __shared__ float red[BATCH];
    const int t = threadIdx.x;
    red[t] = row_lse[t];
    __syncthreads();
    for (int o = BATCH / 2; o > 0; o >>= 1) {
        if (t < o) red[t] += red[t + o];
        __syncthreads();
    }
    if (t == 0) out[0] = red[0] * (1.0f / (float)BATCH);
}

// ---------------------------------------------------------------------------
// Kernel D: 256 MB bank copy with non-temporal load/store (don't thrash L2).
// Grid-stride scalar dwords (dst is d_out+1 -> only 4B aligned).
// ---------------------------------------------------------------------------
__global__ void copy_bank_kernel(const float* __restrict__ bank,
                                 float* __restrict__ out, long long n) {
    long long i = (long long)blockIdx.x * blockDim.x + threadIdx.x;
    const long long stride = (long long)gridDim.x * blockDim.x;
    for (; i < n; i += stride) {
        __builtin_nontemporal_store(__builtin_nontemporal_load(bank + i), out + i);
    }
}

// ---------------------------------------------------------------------------
// Kernel D2: scatter-update 512 rows: normalize(0.5*bank[idx] + 0.5*t_n).
// Reads the ORIGINAL bank (inputs are never mutated).  512 blocks x 128 thr.
// ---------------------------------------------------------------------------
__global__ void scatter_kernel(const float* __restrict__ bank,
                               const float* __restrict__ t_n,
                               const int* __restrict__ indices,
                               float* __restrict__ out_bank) {
    const int b = blockIdx.x;
    const int t = threadIdx.x;
    const long long row = (long long)indices[b];
    __shared__ float red[FEAT_DIM];

    const float u = 0.5f * bank[row * FEAT_DIM + t] + 0.5f * t_n[b * FEAT_DIM + t];
    red[t] = u * u;
    __syncthreads();
    for (int o = FEAT_DIM / 2; o > 0; o >>= 1) {
        if (t < o) red[t] += red[t + o];
        __syncthreads();
    }
    const float n = sqrtf(red[0]);
    out_bank[row * FEAT_DIM + t] = u / fmaxf(n, EPS_N);
}

// ---------------------------------------------------------------------------
extern "C" void kernel_launch(void* const* d_in, const int* in_sizes, int n_in,
                              void* d_out, int out_size, void* d_ws, size_t ws_size,
                              hipStream_t stream) {
    const float* student  = (const float*)d_in[0];
    const float* teacher  = (const float*)d_in[1];
    const float* bank     = (const float*)d_in[2];
    const int*   indices  = (const int*)d_in[3];
    const int*   neg_idx  = (const int*)d_in[4];

    float* out      = (float*)d_out;        // [0] = loss
    float* out_bank = out + 1;              // [1..] = new memory bank

    // workspace layout (floats)
    float* ws       = (float*)d_ws;
    float* s_n      = ws;                                   // 512*128
    float* t_n      = s_n + BATCH * FEAT_DIM;               // 512*128
    float* pos      = t_n + BATCH * FEAT_DIM;               // 512
    float* row_lse  = pos + BATCH;                          // 512
    float* neg_log  = row_lse + BATCH;                      // 512*4096

    normalize_kernel<<<BATCH, FEAT_DIM, 0, stream>>>(student, teacher, s_n, t_n, pos);

    neg_logits_kernel<<<BATCH, 256, 0, stream>>>(bank, neg_idx, s_n, neg_log);

    lse_kernel<<<BATCH, 256, 0, stream>>>(neg_log, pos, row_lse);
    mean_kernel<<<1, BATCH, 0, stream>>>(row_lse, out);

    const long long nbank = (long long)N_DATA * FEAT_DIM;
    copy_bank_kernel<<<8192, 256, 0, stream>>>(bank, out_bank, nbank);

    scatter_kernel<<<BATCH, FEAT_DIM, 0, stream>>>(bank, t_n, indices, out_bank);
}